// MatrixVectorScaledDotProductAttention_8847632630095
// MI455X (gfx1250) — compile-verified
//
#include <hip/hip_runtime.h>

typedef __attribute__((ext_vector_type(2))) float v2f;
typedef __attribute__((ext_vector_type(4))) float v4f;
typedef __attribute__((ext_vector_type(8))) float v8f;

#define NB 512
#define LSEQ 2048
#define DK 128
#define DV 128
#define INV_TEMP (1.0f / 11.313708498984761f)
#define NTHREADS 256
#define NWAVES 8

__global__ __launch_bounds__(NTHREADS) void
mv_sdpa_kernel(const float* __restrict__ q,
               const float* __restrict__ k,
               const float* __restrict__ v,
               float* __restrict__ out,
               float* __restrict__ attn_out)
{
    __shared__ float qs[DK];            // q / temperature
    __shared__ float sc[LSEQ];          // scores -> exp -> attn
    __shared__ float red[NWAVES * DV];  // per-wave output partials
    __shared__ float rtmp[NWAVES];      // cross-wave scalar reduction

    const int b    = blockIdx.x;
    const int tid  = threadIdx.x;
    const int lane = tid & 31;
    const int wave = tid >> 5;

    // ---- load q, pre-scaled by 1/T (equivalent to scaling scores) ----
    if (tid < DK) qs[tid] = q[(size_t)b * DK + tid] * INV_TEMP;
    __syncthreads();

    const float* kb = k + (size_t)b * LSEQ * DK;

    // ---- phase 1: scores[l] = (q/T) . k[l,:]  via V_WMMA_F32_16X16X4_F32 ----
    // A layout (16x4 f32): lanes 0-15 -> M=lane, VGPR{0,1}=K{base+0,base+1};
    //                      lanes 16-31 -> M=lane-16, VGPR{0,1}=K{base+2,base+3}.
    // B is q replicated across all 16 columns with the mirrored K mapping,
    // so every D column holds the same 16 scores.
    const int row  = lane & 15;
    const int koff = (lane >> 4) * 2;  // 0 for lanes 0-15, 2 for lanes 16-31

    for (int tile = wave; tile < LSEQ / 16; tile += NWAVES) {
        const int l0 = tile * 16;
        const float* krow = kb + (size_t)(l0 + row) * DK + koff;

        // keep the HBM/L2 stream ahead: touch this wave's next tile (l0+128)
        if (l0 + 128 < LSEQ)
            __builtin_prefetch(kb + (size_t)(l0 + 128 + row) * DK + koff, 0, 1);

        v8f acc = {};
#pragma unroll 8
        for (int kc = 0; kc < DK; kc += 4) {
            v2f a = *(const v2f*)(krow + kc);   // global_load_b64 per lane
            v2f bq;
            bq.x = qs[kc + koff];
            bq.y = qs[kc + koff + 1];
            acc = __builtin_amdgcn_wmma_f32_16x16x4_f32(
                false, a, false, bq, (short)0, acc, false, false);
        }
        // D layout: VGPR j -> M=j (lanes 0-15) / M=j+8 (lanes 16-31), N=lane%16.
        // Columns identical -> lanes 0 and 16 write the 16 scores.
        if (row == 0) {
            const int mbase = l0 + ((lane >> 4) ? 8 : 0);
#pragma unroll
            for (int j = 0; j < 8; ++j) sc[mbase + j] = acc[j];
        }
    }
    __syncthreads();

    // ---- phase 2: softmax over L (block-wide, scores live in LDS) ----
    float m = -INFINITY;
    for (int l = tid; l < LSEQ; l += NTHREADS) m = fmaxf(m, sc[l]);
#pragma unroll
    for (int off = 16; off > 0; off >>= 1) m = fmaxf(m, __shfl_xor(m, off, 32));
    if (lane == 0) rtmp[wave] = m;
    __syncthreads();
    if (tid == 0) {
        float mm = rtmp[0];
#pragma unroll
        for (int i = 1; i < NWAVES; ++i) mm = fmaxf(mm, rtmp[i]);
        rtmp[0] = mm;
    }
    __syncthreads();
    m = rtmp[0];
    __syncthreads();  // everyone has read rtmp[0] before it is reused

    float s = 0.0f;
    for (int l = tid; l < LSEQ; l += NTHREADS) {
        float e = __expf(sc[l] - m);
        sc[l] = e;
        s += e;
    }
#pragma unroll
    for (int off = 16; off > 0; off >>= 1) s += __shfl_xor(s, off, 32);
    if (lane == 0) rtmp[wave] = s;
    __syncthreads();
    if (tid == 0) {
        float ss = 0.0f;
#pragma unroll
        for (int i = 0; i < NWAVES; ++i) ss += rtmp[i];
        rtmp[0] = ss;
    }
    __syncthreads();
    const float inv_sum = 1.0f / rtmp[0];

    for (int l = tid; l < LSEQ; l += NTHREADS) {
        float a = sc[l] * inv_sum;
        sc[l] = a;
        attn_out[(size_t)b * LSEQ + l] = a;  // second reference output
    }
    __syncthreads();

    // ---- phase 3: out[d] = sum_l attn[l] * v[l,d] ----
    // One wave spans all 128 dims with float4 per lane (coalesced b128 loads);
    // waves stride L by 8; cross-wave reduction through LDS.
    const float* vb = v + (size_t)b * LSEQ * DV;
    const int d4 = lane * 4;
    v4f accv = {};
    for (int l = wave; l < LSEQ; l += NWAVES) {
        float a = sc[l];
        v4f vv = *(const v4f*)(vb + (size_t)l * DV + d4);  // global_load_b128
        accv += vv * a;
    }
    *(v4f*)(&red[wave * DV + d4]) = accv;
    __syncthreads();

    if (tid < DV) {
        float o = 0.0f;
#pragma unroll
        for (int w = 0; w < NWAVES; ++w) o += red[w * DV + tid];
        out[(size_t)b * DV + tid] = o;
    }
}

extern "C" void kernel_launch(void* const* d_in, const int* in_sizes, int n_in,
                              void* d_out, int out_size, void* d_ws, size_t ws_size,
                              hipStream_t stream) {
    (void)in_sizes; (void)n_in; (void)out_size; (void)d_ws; (void)ws_size;
    const float* q = (const float*)d_in[0];
    const float* k = (const float*)d_in[1];
    const float* v = (const float*)d_in[2];
    float* out  = (float*)d_out;                       // (512,128)
    float* attn = out + (size_t)NB * DV;               // (512,2048)
    mv_sdpa_kernel<<<NB, NTHREADS, 0, stream>>>(q, k, v, out, attn);
}